// PlacementOnlyNoBinNetwork_12884901888400
// MI455X (gfx1250) — compile-verified
//
#include <hip/hip_runtime.h>
#include <math.h>

typedef __attribute__((ext_vector_type(2))) float v2f;
typedef __attribute__((ext_vector_type(8))) float v8f;

#define B_   256
#define S_   128
#define D_   64
#define H_   512
#define G4H_ 2048

__device__ __forceinline__ float sigmoidf_(float x) { return 1.0f / (1.0f + __expf(-x)); }

// ---------------------------------------------------------------------------
// C[M,N] = A1[M,K1] @ W1[N,K1]^T  (+ A2[M,K2] @ W2[N,K2]^T if K2>0)
// fp32 WMMA 16x16x4. One wave per 16x64 output strip (4 accumulators).
// A fragment (16x4, MxK): lanes 0-15 -> A[row, k..k+1]; lanes 16-31 -> A[row, k+2..k+3]
// B fragment (4x16, KxN): lanes 0-15 -> W[col, k..k+1]; lanes 16-31 -> W[col, k+2..k+3]
// ---------------------------------------------------------------------------
__global__ void __launch_bounds__(128)
gemm_wmma_f32(const float* __restrict__ A1, int lda1,
              const float* __restrict__ W1, int K1,
              const float* __restrict__ A2, int lda2,
              const float* __restrict__ W2, int K2,
              float* __restrict__ C, int M, int N)
{
    const int lane = threadIdx.x & 31;
    const int wave = threadIdx.x >> 5;
    const int l15  = lane & 15;
    const int half = lane >> 4;            // 0 or 1
    const int koff = half << 1;            // 0 or 2

    const int mTiles = M >> 4;
    const int gw = blockIdx.x * 4 + wave;
    const int mt = gw % mTiles;
    const int ns = gw / mTiles;

    const int row  = (mt << 4) + l15;      // A row for this lane
    const int colb = ns * 64 + l15;        // base B column for this lane

    v8f acc0 = {}, acc1 = {}, acc2 = {}, acc3 = {};

    {
        const float* Ar = A1 + (size_t)row * lda1 + koff;
        const float* Wp0 = W1 + (size_t)(colb +  0) * K1 + koff;
        const float* Wp1 = W1 + (size_t)(colb + 16) * K1 + koff;
        const float* Wp2 = W1 + (size_t)(colb + 32) * K1 + koff;
        const float* Wp3 = W1 + (size_t)(colb + 48) * K1 + koff;
        for (int k = 0; k < K1; k += 4) {
            v2f a  = *(const v2f*)(Ar  + k);
            v2f b0 = *(const v2f*)(Wp0 + k);
            v2f b1 = *(const v2f*)(Wp1 + k);
            v2f b2 = *(const v2f*)(Wp2 + k);
            v2f b3 = *(const v2f*)(Wp3 + k);
            acc0 = __builtin_amdgcn_wmma_f32_16x16x4_f32(false, a, false, b0, (short)0, acc0, false, false);
            acc1 = __builtin_amdgcn_wmma_f32_16x16x4_f32(false, a, false, b1, (short)0, acc1, false, false);
            acc2 = __builtin_amdgcn_wmma_f32_16x16x4_f32(false, a, false, b2, (short)0, acc2, false, false);
            acc3 = __builtin_amdgcn_wmma_f32_16x16x4_f32(false, a, false, b3, (short)0, acc3, false, false);
        }
    }
    if (K2 > 0) {
        const float* Ar = A2 + (size_t)row * lda2 + koff;
        const float* Wp0 = W2 + (size_t)(colb +  0) * K2 + koff;
        const float* Wp1 = W2 + (size_t)(colb + 16) * K2 + koff;
        const float* Wp2 = W2 + (size_t)(colb + 32) * K2 + koff;
        const float* Wp3 = W2 + (size_t)(colb + 48) * K2 + koff;
        for (int k = 0; k < K2; k += 4) {
            v2f a  = *(const v2f*)(Ar  + k);
            v2f b0 = *(const v2f*)(Wp0 + k);
            v2f b1 = *(const v2f*)(Wp1 + k);
            v2f b2 = *(const v2f*)(Wp2 + k);
            v2f b3 = *(const v2f*)(Wp3 + k);
            acc0 = __builtin_amdgcn_wmma_f32_16x16x4_f32(false, a, false, b0, (short)0, acc0, false, false);
            acc1 = __builtin_amdgcn_wmma_f32_16x16x4_f32(false, a, false, b1, (short)0, acc1, false, false);
            acc2 = __builtin_amdgcn_wmma_f32_16x16x4_f32(false, a, false, b2, (short)0, acc2, false, false);
            acc3 = __builtin_amdgcn_wmma_f32_16x16x4_f32(false, a, false, b3, (short)0, acc3, false, false);
        }
    }

    // C/D layout: VGPR r, lanes 0-15 -> M=base+r, lanes 16-31 -> M=base+8+r
    const int r0 = (mt << 4) + (half << 3);
    const int c0 = ns * 64 + l15;
    #pragma unroll
    for (int r = 0; r < 8; r++) {
        float* cr = C + (size_t)(r0 + r) * N + c0;
        cr[ 0] = acc0[r];
        cr[16] = acc1[r];
        cr[32] = acc2[r];
        cr[48] = acc3[r];
    }
}

// ---------------------------------------------------------------------------
// LSTM pointwise: z = Z + bias, gate order i,f,g,o along 4H axis
// ---------------------------------------------------------------------------
__global__ void lstm_pointwise(const float* __restrict__ Z, const float* __restrict__ bias,
                               float* __restrict__ h, float* __restrict__ c,
                               float* __restrict__ enc_out, int t)
{
    int idx = blockIdx.x * blockDim.x + threadIdx.x;   // B*H threads
    int b = idx >> 9;
    int u = idx & (H_ - 1);
    const float* zb = Z + (size_t)b * G4H_;
    float zi = zb[u]            + bias[u];
    float zf = zb[H_ + u]       + bias[H_ + u];
    float zg = zb[2 * H_ + u]   + bias[2 * H_ + u];
    float zo = zb[3 * H_ + u]   + bias[3 * H_ + u];
    float cn = sigmoidf_(zf) * c[idx] + sigmoidf_(zi) * tanhf(zg);
    float hn = sigmoidf_(zo) * tanhf(cn);
    c[idx] = cn;
    h[idx] = hn;
    if (enc_out) enc_out[((size_t)b * S_ + t) * H_ + u] = hn;
}

// ---------------------------------------------------------------------------
// scores[b,s] = dot(h[b,:], enc_out[b,s,:])   one wave32 per score
// ---------------------------------------------------------------------------
__global__ void attn_scores(const float* __restrict__ h, const float* __restrict__ enc_out,
                            float* __restrict__ scores)
{
    int gw   = blockIdx.x * 8 + (threadIdx.x >> 5);
    int lane = threadIdx.x & 31;
    int b = gw >> 7;
    int s = gw & (S_ - 1);
    const float* e  = enc_out + ((size_t)b * S_ + s) * H_;
    const float* hv = h + (size_t)b * H_;
    float acc = 0.f;
    #pragma unroll 4
    for (int i = 0; i < H_; i += 32) acc += e[i + lane] * hv[i + lane];
    for (int off = 16; off > 0; off >>= 1) acc += __shfl_xor(acc, off, 32);
    if (lane == 0) scores[gw] = acc;
}

// ---------------------------------------------------------------------------
// per batch element: softmax over S, logits[j] = sum_s attn[s]*P[b,s,j] + dense_b[j],
// apply mask (-inf), write out[:,t,:], then update mask with tgt[:,t]
// ---------------------------------------------------------------------------
__global__ void __launch_bounds__(128)
attn_logits(const float* __restrict__ scores, const float* __restrict__ P,
            const float* __restrict__ dense_b, int* __restrict__ mask,
            const int* __restrict__ tgt, float* __restrict__ out, int t)
{
    __shared__ float attn_s[S_];
    __shared__ float red[S_];
    int b = blockIdx.x, j = threadIdx.x;

    float sc = scores[b * S_ + j];
    red[j] = sc; __syncthreads();
    for (int off = 64; off > 0; off >>= 1) {
        if (j < off) red[j] = fmaxf(red[j], red[j + off]);
        __syncthreads();
    }
    float mx = red[0]; __syncthreads();
    float e = __expf(sc - mx);
    red[j] = e; __syncthreads();
    for (int off = 64; off > 0; off >>= 1) {
        if (j < off) red[j] += red[j + off];
        __syncthreads();
    }
    attn_s[j] = e * (1.0f / red[0]);
    __syncthreads();

    float acc = dense_b[j];
    const float* Pb = P + (size_t)b * S_ * S_ + j;
    #pragma unroll 4
    for (int s = 0; s < S_; s++) acc += attn_s[s] * Pb[(size_t)s * S_];

    out[((size_t)b * S_ + t) * S_ + j] = mask[b * S_ + j] ? -__builtin_inff() : acc;
    __syncthreads();
    if (j == 0) {
        int tg = tgt[b * S_ + t];
        if (tg >= 0 && tg < S_) mask[b * S_ + tg] = 1;
    }
}

// dec_x[b,t,:] = (t==0) ? 0 : parts[b, clip(tgt[b,t-1]), :]
__global__ void gather_decx(const float* __restrict__ parts, const int* __restrict__ tgt,
                            float* __restrict__ dec_x)
{
    int idx = blockIdx.x * blockDim.x + threadIdx.x;   // B*S*D
    int d = idx & (D_ - 1);
    int t = (idx >> 6) & (S_ - 1);
    int b = idx >> 13;
    float v = 0.f;
    if (t > 0) {
        int tg = tgt[b * S_ + t - 1];
        tg = tg < 0 ? 0 : (tg > S_ - 1 ? S_ - 1 : tg);
        v = parts[((size_t)b * S_ + tg) * D_ + d];
    }
    dec_x[idx] = v;
}

__global__ void zero_f32(float* __restrict__ p, int n)
{
    int i = blockIdx.x * 256 + threadIdx.x;
    if (i < n) p[i] = 0.f;
}

extern "C" void kernel_launch(void* const* d_in, const int* in_sizes, int n_in,
                              void* d_out, int out_size, void* d_ws, size_t ws_size,
                              hipStream_t stream)
{
    const float* parts   = (const float*)d_in[0];
    // d_in[1] = bin_info (unused by reference)
    const int*   tgt     = (const int*)  d_in[2];
    const float* enc_wih = (const float*)d_in[3];
    const float* enc_whh = (const float*)d_in[4];
    const float* enc_b   = (const float*)d_in[5];
    const float* dec_wih = (const float*)d_in[6];
    const float* dec_whh = (const float*)d_in[7];
    const float* dec_b   = (const float*)d_in[8];
    const float* dense_w = (const float*)d_in[9];
    const float* dense_b = (const float*)d_in[10];
    float* out = (float*)d_out;

    // workspace layout (floats)
    float* ws = (float*)d_ws;
    size_t off = 0;
    float* enc_out = ws + off; off += (size_t)B_ * S_ * H_;     // 16.78M
    float* P       = ws + off; off += (size_t)B_ * S_ * S_;     //  4.19M
    float* Z       = ws + off; off += (size_t)B_ * G4H_;        //  0.52M
    float* hc      = ws + off; off += (size_t)4 * B_ * H_;      //  0.52M (h_enc,c_enc,h_dec,c_dec)
    int*   mask    = (int*)(ws + off); off += (size_t)B_ * S_;  //  0.03M
    float* dec_x   = ws + off; off += (size_t)B_ * S_ * D_;     //  2.10M
    float* scores  = ws + off; off += (size_t)B_ * S_;          //  0.03M

    float* h_enc = hc;
    float* c_enc = hc + (size_t)B_ * H_;
    float* h_dec = hc + (size_t)2 * B_ * H_;
    float* c_dec = hc + (size_t)3 * B_ * H_;

    // zero h/c states + mask (contiguous)
    int nz = 4 * B_ * H_ + B_ * S_;
    zero_f32<<<(nz + 255) / 256, 256, 0, stream>>>(hc, nz);
    gather_decx<<<(B_ * S_ * D_) / 256, 256, 0, stream>>>(parts, tgt, dec_x);

    const int gatesBlocks = ((B_ / 16) * (G4H_ / 64)) / 4;       // 128
    const int pwBlocks    = (B_ * H_) / 256;                     // 512
    const int scBlocks    = (B_ * S_) / 8;                       // 4096

    // ---------------- encoder: 128 sequential LSTM steps ----------------
    for (int t = 0; t < S_; t++) {
        gemm_wmma_f32<<<gatesBlocks, 128, 0, stream>>>(
            h_enc, H_, enc_whh, H_,
            parts + (size_t)t * D_, S_ * D_, enc_wih, D_,
            Z, B_, G4H_);
        lstm_pointwise<<<pwBlocks, 256, 0, stream>>>(Z, enc_b, h_enc, c_enc, enc_out, t);
    }

    // ---------------- P = enc_out @ dense_w^T  [B*S, S] ----------------
    const int pBlocks = (((B_ * S_) / 16) * (S_ / 64)) / 4;      // 1024
    gemm_wmma_f32<<<pBlocks, 128, 0, stream>>>(
        enc_out, H_, dense_w, H_, nullptr, 0, nullptr, 0, P, B_ * S_, S_);

    // ---------------- decoder: 128 sequential steps ----------------
    for (int t = 0; t < S_; t++) {
        gemm_wmma_f32<<<gatesBlocks, 128, 0, stream>>>(
            h_dec, H_, dec_whh, H_,
            dec_x + (size_t)t * D_, S_ * D_, dec_wih, D_,
            Z, B_, G4H_);
        lstm_pointwise<<<pwBlocks, 256, 0, stream>>>(Z, dec_b, h_dec, c_dec, nullptr, t);
        attn_scores<<<scBlocks, 256, 0, stream>>>(h_dec, enc_out, scores);
        attn_logits<<<B_, 128, 0, stream>>>(scores, P, dense_b, mask, tgt, out, t);
    }
}